// LatentSpace_22471268892719
// MI455X (gfx1250) — compile-verified
//
#include <hip/hip_runtime.h>
#include <math.h>

// ---------------------------------------------------------------------------
// LatentSpace log_p_z, fused for MI455X (gfx1250, wave32, WMMA).
//
//   prep_frag   : packs W = [a1 | mu_B] (64x288, zero-padded) directly into
//                 WMMA B-fragment order: one float4 per (tile, k-pair, lane)
//                 so the GEMM kernel streams coalesced b128 loads.
//   prep_cst    : per-state constants inv_sig, recip, a1.muB, ||muB||^2,
//                 log softmax(pi).
//   gmm_kernel  : one wave per 16 z-rows. f32 WMMA 16x16x4 GEMM
//                 Z(16x64) x W(64x288) fused with all elementwise math
//                 (ndtr/log) and online logsumexp over the 136 states.
//   finish      : deterministic fixed-order sum of per-block partials, / N.
// ---------------------------------------------------------------------------

typedef __attribute__((ext_vector_type(2))) float v2f;
typedef __attribute__((ext_vector_type(4))) float v4f;
typedef __attribute__((ext_vector_type(8))) float v8f;

#define DIM     64
#define NCLU    16
#define NST     136
#define NPAD    144          // states padded to 9 tiles of 16
#define NTILES  18           // 9 a1-tiles + 9 mu_B-tiles
#define LOG2PI  1.8378770664093453f

// Fragment store: Wf[(n*8 + kk)*32 + lane] = float4
//   {B[kb][col], B[kb+1][col], B[kb+4][col], B[kb+5][col]}
// where kb = 8*kk + (lane>=16 ? 2 : 0), col = 16*n + (lane&15).
// This is exactly the per-lane data for WMMA k-steps 2kk and 2kk+1.

// ---------------------------------------------------------------------------
__global__ void prep_frag(const float* __restrict__ mu,   // (64,16) row-major
                          const int*   __restrict__ Aidx,
                          const int*   __restrict__ Bidx,
                          v4f* __restrict__ Wf)
{
    const int n    = blockIdx.x;            // 0..17
    const int kk   = threadIdx.x >> 5;      // 0..7
    const int lane = threadIdx.x & 31;
    const bool hi  = lane >= 16;
    const int  c   = lane & 15;

    const int  col  = 16 * n + c;           // 0..287
    const bool isA1 = col < NPAD;
    const int  s    = isA1 ? col : (col - NPAD);
    const int  d0   = 8 * kk + (hi ? 2 : 0);

    v4f q = {0.f, 0.f, 0.f, 0.f};
    if (s < NST) {
        const int ca = Aidx[s], cb = Bidx[s];
        const int dd[4] = {d0, d0 + 1, d0 + 4, d0 + 5};
        float v[4];
#pragma unroll
        for (int i = 0; i < 4; ++i) {
            const float mb = mu[dd[i] * NCLU + cb];
            v[i] = isA1 ? (mb - mu[dd[i] * NCLU + ca]) : mb;
        }
        q.x = v[0]; q.y = v[1]; q.z = v[2]; q.w = v[3];
    }
    Wf[(n * 8 + kk) * 32 + lane] = q;
}

// ---------------------------------------------------------------------------
__global__ void prep_cst(const float* __restrict__ mu,
                         const float* __restrict__ pi,
                         const int*   __restrict__ Aidx,
                         const int*   __restrict__ Bidx,
                         float* __restrict__ cst)        // 5*144 floats
{
    const int t = threadIdx.x;
    if (t >= NST) return;

    const int ca = Aidx[t], cb = Bidx[t];
    float inv = 0.f, c1 = 0.f, nb = 0.f;
    for (int d = 0; d < DIM; ++d) {
        const float ma = mu[d * NCLU + ca];
        const float mb = mu[d * NCLU + cb];
        const float a1 = mb - ma;
        inv = fmaf(a1, a1, inv);
        c1  = fmaf(a1, mb, c1);
        nb  = fmaf(mb, mb, nb);
    }
    // log(softmax(pi) + 1e-12)
    float mx = -INFINITY;
    for (int j = 0; j < NST; ++j) mx = fmaxf(mx, pi[j]);
    float se = 0.f;
    for (int j = 0; j < NST; ++j) se += expf(pi[j] - mx);
    const float lp = logf(expf(pi[t] - mx) / se + 1e-12f);

    cst[t]            = inv;
    cst[NPAD + t]     = (inv == 0.0f) ? 0.0f : 1.0f / inv;
    cst[2 * NPAD + t] = c1;
    cst[3 * NPAD + t] = nb;
    cst[4 * NPAD + t] = lp;
}

// ---------------------------------------------------------------------------
__global__ void __launch_bounds__(32)
gmm_kernel(const float* __restrict__ Z,     // (rows, 64)
           const v4f*   __restrict__ Wf,    // packed B fragments
           const float* __restrict__ cst,   // 5*144
           float* __restrict__ partial)     // per-block row sums of lse
{
    const int  lane = threadIdx.x;          // wave32
    const bool hi   = lane >= 16;
    const int  c    = lane & 15;
    const int  m0   = blockIdx.x * 16;

    // ---- A fragments: 16 K-steps of the 16x4 f32 tile of Z ----------------
    // lanes 0-15: (K=4k, 4k+1) of row m0+lane; lanes 16-31: (K=4k+2, 4k+3).
    const float* zrow = Z + (size_t)(m0 + c) * DIM + (hi ? 2 : 0);
    v2f   a[16];
    float pn = 0.f;                          // partial ||z||^2 from fragments
#pragma unroll
    for (int k = 0; k < 16; ++k) {
        const v2f av = *reinterpret_cast<const v2f*>(zrow + 4 * k);
        a[k] = av;
        pn = fmaf(av.x, av.x, pn);
        pn = fmaf(av.y, av.y, pn);
    }
    const float fullnorm = pn + __shfl_xor(pn, 16);   // ||z[m0 + (lane&15)]||^2
    float zn[8];                                       // ||z||^2 for my 8 rows
#pragma unroll
    for (int r = 0; r < 8; ++r)
        zn[r] = __shfl(fullnorm, r + (hi ? 8 : 0));

    // ---- online logsumexp state per C-row ---------------------------------
    float M[8], S[8];
#pragma unroll
    for (int r = 0; r < 8; ++r) { M[r] = -INFINITY; S[r] = 0.f; }

    const float base = -0.5f * (float)DIM * LOG2PI;

    for (int n = 0; n < 9; ++n) {            // 9 tile pairs: dot1 & dot2
        v8f d1 = {};
        v8f d2 = {};
#pragma unroll
        for (int kk = 0; kk < 8; ++kk) {     // two k-steps per b128 load
            const v4f q1 = Wf[((n    ) * 8 + kk) * 32 + lane];
            const v4f q2 = Wf[((n + 9) * 8 + kk) * 32 + lane];
            v2f b;
            b.x = q1.x; b.y = q1.y;
            d1 = __builtin_amdgcn_wmma_f32_16x16x4_f32(
                     false, a[2 * kk], false, b, (short)0, d1, false, false);
            b.x = q2.x; b.y = q2.y;
            d2 = __builtin_amdgcn_wmma_f32_16x16x4_f32(
                     false, a[2 * kk], false, b, (short)0, d2, false, false);
            b.x = q1.z; b.y = q1.w;
            d1 = __builtin_amdgcn_wmma_f32_16x16x4_f32(
                     false, a[2 * kk + 1], false, b, (short)0, d1, false, false);
            b.x = q2.z; b.y = q2.w;
            d2 = __builtin_amdgcn_wmma_f32_16x16x4_f32(
                     false, a[2 * kk + 1], false, b, (short)0, d2, false, false);
        }

        const int s = 16 * n + c;
        if (s < NST) {
            const float inv   = cst[s];
            const float rc    = cst[NPAD + s];
            const float c1    = cst[2 * NPAD + s];
            const float nb    = cst[3 * NPAD + s];
            const float lp    = cst[4 * NPAD + s];
            const float sd    = sqrtf(inv + 1e-12f);
            const float lginv = logf(inv + 1e-12f);
#pragma unroll
            for (int r = 0; r < 8; ++r) {
                const float dot1 = d1[r];            // z . a1[:,s]
                const float dot2 = d2[r];            // z . mu_B[:,s]
                const float sqa2 = zn[r] - 2.f * dot2 + nb;
                float t;
                if (inv == 0.0f) {
                    t = -0.5f * sqa2;
                } else {
                    const float muu = -(dot1 - c1) * rc;
                    const float _t  = fmaf(muu * muu, inv, -sqa2);
                    const float cd  = normcdff((1.f - muu) * sd)
                                    - normcdff(-muu * sd);
                    t = 0.5f * (LOG2PI - lginv + _t) + logf(cd + 1e-12f);
                }
                t += lp + base;
                // online logsumexp update
                const float mn = fmaxf(M[r], t);
                S[r] = S[r] * expf(M[r] - mn) + expf(t - mn);
                M[r] = mn;
            }
        }
    }

    // ---- combine (M,S) across the 16-lane half, then sum rows -------------
    float rowsum = 0.f;
#pragma unroll
    for (int r = 0; r < 8; ++r) {
        float m = M[r], sv = S[r];
#pragma unroll
        for (int msk = 1; msk <= 8; msk <<= 1) {   // stays within 16-lane half
            const float mo = __shfl_xor(m, msk);
            const float so = __shfl_xor(sv, msk);
            const float mn = fmaxf(m, mo);
            sv = sv * expf(m - mn) + so * expf(mo - mn);
            m  = mn;
        }
        rowsum += m + logf(sv);                    // lse of row m0+r(+8)
    }
    const float tot = rowsum + __shfl_xor(rowsum, 16);   // all 16 rows
    if (lane == 0) partial[blockIdx.x] = tot;
}

// ---------------------------------------------------------------------------
__global__ void finish_kernel(const float* __restrict__ partial, int nblocks,
                              int nrows, float* __restrict__ out)
{
    if (threadIdx.x == 0 && blockIdx.x == 0) {
        float s = 0.f;
        for (int i = 0; i < nblocks; ++i) s += partial[i];   // fixed order
        out[0] = s / (float)nrows;
    }
}

// ---------------------------------------------------------------------------
extern "C" void kernel_launch(void* const* d_in, const int* in_sizes, int n_in,
                              void* d_out, int out_size, void* d_ws, size_t ws_size,
                              hipStream_t stream)
{
    const float* z  = (const float*)d_in[0];   // (512,10,64)
    const float* mu = (const float*)d_in[1];   // (64,16)
    const float* pi = (const float*)d_in[2];   // (1,136)
    const int*   A  = (const int*)d_in[3];     // (136,)
    const int*   B  = (const int*)d_in[4];     // (136,)
    float*       out = (float*)d_out;

    float* ws      = (float*)d_ws;
    v4f*   Wf      = (v4f*)ws;                  // 18*8*32 float4 = 18432 floats
    float* cst     = ws + NTILES * 8 * 32 * 4;  // 5*144 floats
    float* partial = cst + 5 * NPAD;            // nblocks floats

    const int rows    = in_sizes[0] / DIM;      // 5120
    const int nblocks = rows / 16;              // 320

    prep_frag<<<NTILES, 256, 0, stream>>>(mu, A, B, Wf);
    prep_cst<<<1, 160, 0, stream>>>(mu, pi, A, B, cst);
    gmm_kernel<<<nblocks, 32, 0, stream>>>(z, Wf, cst, partial);
    finish_kernel<<<1, 32, 0, stream>>>(partial, nblocks, rows, out);
}